// MultiheadA_9474697855664
// MI455X (gfx1250) — compile-verified
//
#include <hip/hip_runtime.h>
#include <cstdint>

// ---------------------------------------------------------------------------
// MI455X (gfx1250) multihead attention fwd, fp32 end-to-end.
// Memory-bound on the 537MB qk logit writeout -> single-pass flash attention.
// V_WMMA_F32_16X16X4_F32 everywhere: compute (~70 GFLOP) is far below the
// matrix ceiling while HBM traffic (~0.6 GB) bounds runtime, so fp32 WMMA
// keeps exact reference precision for free. GEMM waves own 32x32 register
// tiles (2x2 WMMA accumulators) to halve LDS fragment traffic, and tile
// staging uses GLOBAL_LOAD_ASYNC_TO_LDS_B128 (ASYNCcnt) where no transpose
// is needed.
// ---------------------------------------------------------------------------

#define BATCH 2
#define SEQ   2048
#define DIM   1024
#define HEADS 16
#define HD    64

typedef float v2f __attribute__((ext_vector_type(2)));
typedef float v8f __attribute__((ext_vector_type(8)));
typedef int   i128v __attribute__((vector_size(16)));   // matches builtin param

// ---------------- async global->LDS staging (CDNA5 ASYNCcnt path) ----------
#if defined(__gfx1250__) && __has_builtin(__builtin_amdgcn_global_load_async_to_lds_b128)
#define USE_ASYNC_LDS 1
#endif

__device__ __forceinline__ void stage_b128(const float* g, float* l) {
#ifdef USE_ASYNC_LDS
  __builtin_amdgcn_global_load_async_to_lds_b128(
      (__attribute__((address_space(1))) i128v*)(uintptr_t)g,
      (__attribute__((address_space(3))) i128v*)(uintptr_t)l,
      0, 0);
#else
  *(float4*)l = *(const float4*)g;
#endif
}

__device__ __forceinline__ void stage_wait() {
#ifdef USE_ASYNC_LDS
#if __has_builtin(__builtin_amdgcn_s_wait_asynccnt)
  __builtin_amdgcn_s_wait_asynccnt(0);
#else
  asm volatile("s_wait_asynccnt 0x0" ::: "memory");
#endif
#endif
}

// ============================ GEMM: Out = X @ W^T (+bias) ==================
// X: MxK row-major, W: NxK row-major. Block tile 128(M) x 64(N), K chunk 32.
// 8 waves as 4(M) x 2(N); each wave a 32x32 register tile = 2x2 WMMA C tiles,
// so every {a0,a1,b0,b1} fragment fetch feeds 4 WMMAs.
#define GM 128
#define GN 64
#define GK 32

__global__ __launch_bounds__(256) void gemm_xwt_kernel(
    const float* __restrict__ X, const float* __restrict__ W,
    const float* __restrict__ bias, float* __restrict__ Out,
    int M, int N, int K)
{
  __shared__ __align__(16) float As[GM][GK + 8];   // 128 x 40 (160B rows)
  __shared__ __align__(16) float Bs[GK][GN + 1];   // Bs[k][n] = W[n][k]

  const int tid  = threadIdx.x;
  const int wave = tid >> 5;
  const int lane = tid & 31;
  const int ln   = lane & 15;      // fragment M-row / N-col selector
  const int grp  = lane >> 4;      // half-wave: K pair (A/B), M+8 (C)
  const int m0   = (wave & 3) * 32;
  const int n0   = (wave >> 2) * 32;
  const int blockM = blockIdx.y * GM;
  const int blockN = blockIdx.x * GN;

  v8f acc[2][2];
#pragma unroll
  for (int i = 0; i < 2; ++i)
#pragma unroll
    for (int j = 0; j < 2; ++j) acc[i][j] = (v8f){};

  for (int k0 = 0; k0 < K; k0 += GK) {
    __syncthreads();
    // stage A: 128x32 fp32 = 1024 float4, 4 per thread, straight copy -> async
#pragma unroll
    for (int i = 0; i < 4; ++i) {
      int flat = i * 256 + tid;
      int r = flat >> 3, c4 = flat & 7;
      stage_b128(&X[(size_t)(blockM + r) * K + k0 + c4 * 4], &As[r][c4 * 4]);
    }
    // stage B: W tile 64(n) x 32(k), transposed -> through VGPRs
#pragma unroll
    for (int i = 0; i < 2; ++i) {
      int flat = i * 256 + tid;
      int r = flat >> 3, c4 = flat & 7;
      float4 v = *(const float4*)&W[(size_t)(blockN + r) * K + k0 + c4 * 4];
      Bs[c4 * 4 + 0][r] = v.x; Bs[c4 * 4 + 1][r] = v.y;
      Bs[c4 * 4 + 2][r] = v.z; Bs[c4 * 4 + 3][r] = v.w;
    }
    if (k0 + GK < K) {
      __builtin_prefetch(&X[(size_t)(blockM + (tid >> 1)) * K + k0 + GK], 0, 3);
      __builtin_prefetch(&W[(size_t)(blockN + (tid >> 2)) * K + k0 + GK], 0, 3);
    }
    stage_wait();
    __syncthreads();

#pragma unroll
    for (int kk = 0; kk < GK; kk += 4) {
      v2f a0, a1, b0, b1;
      a0.x = As[m0 + ln][kk + 2 * grp];       a0.y = As[m0 + ln][kk + 2 * grp + 1];
      a1.x = As[m0 + 16 + ln][kk + 2 * grp];  a1.y = As[m0 + 16 + ln][kk + 2 * grp + 1];
      b0.x = Bs[kk + 2 * grp][n0 + ln];       b0.y = Bs[kk + 2 * grp + 1][n0 + ln];
      b1.x = Bs[kk + 2 * grp][n0 + 16 + ln];  b1.y = Bs[kk + 2 * grp + 1][n0 + 16 + ln];
      acc[0][0] = __builtin_amdgcn_wmma_f32_16x16x4_f32(false, a0, false, b0, (short)0, acc[0][0], false, false);
      acc[0][1] = __builtin_amdgcn_wmma_f32_16x16x4_f32(false, a0, false, b1, (short)0, acc[0][1], false, false);
      acc[1][0] = __builtin_amdgcn_wmma_f32_16x16x4_f32(false, a1, false, b0, (short)0, acc[1][0], false, false);
      acc[1][1] = __builtin_amdgcn_wmma_f32_16x16x4_f32(false, a1, false, b1, (short)0, acc[1][1], false, false);
    }
  }

  float bval[2] = {0.0f, 0.0f};
  if (bias) {
    bval[0] = bias[blockN + n0 + ln];
    bval[1] = bias[blockN + n0 + 16 + ln];
  }
#pragma unroll
  for (int i = 0; i < 2; ++i)
#pragma unroll
    for (int j = 0; j < 2; ++j)
#pragma unroll
      for (int r = 0; r < 8; ++r) {
        int row = blockM + m0 + i * 16 + grp * 8 + r;
        Out[(size_t)row * N + blockN + n0 + j * 16 + ln] = acc[i][j][r] + bval[j];
      }
}

// ================================ RoPE =====================================
__global__ __launch_bounds__(256) void rope_kernel(float* __restrict__ T, int total_pairs)
{
  int idx = blockIdx.x * 256 + threadIdx.x;
  if (idx >= total_pairs) return;
  int p    = idx & 31;                 // freq index 0..31
  int rest = idx >> 5;                 // ((b*S + s)*H + h)
  int h    = rest & (HEADS - 1);
  int s    = (rest >> 4) & (SEQ - 1);
  int b    = rest >> 15;
  float freq = (float)s * __expf(-(float)p * (9.210340371976184f / 32.0f));
  float c = __cosf(freq), sn = __sinf(freq);
  size_t base = ((size_t)(b * SEQ + s)) * DIM + h * HD + 2 * p;
  float x1 = T[base], x2 = T[base + 1];
  T[base]     = x1 * c - x2 * sn;
  T[base + 1] = x1 * sn + x2 * c;
}

// ======================= Flash attention + logit writeout ==================
// One WG per (b, h, 128 q-rows). Each wave privately owns a 16-row q strip
// and a 32-key tile -> online-softmax stats stay wave-local (shfl_xor within
// the 16-lane C-fragment halves). Masked logits are 0.0 (NOT -inf) and stay
// in the softmax, matching the reference's multiplicative causal mask.
#define AQB 128
#define AKB 32

__global__ __launch_bounds__(256) void attn_kernel(
    const float* __restrict__ Q, const float* __restrict__ K,
    const float* __restrict__ V, const float* __restrict__ factor,
    float* __restrict__ qk_out, float* __restrict__ attn_out)
{
  __shared__ __align__(16) float Qs[AQB][HD + 4];        // 128 x 68
  __shared__ __align__(16) float Ks[HD][AKB + 2];        // K^T tile: [d][key]
  __shared__ __align__(16) float Vs[AKB][HD + 4];        // [key][d]
  __shared__ __align__(16) float Ps[8][16][AKB + 2];     // wave-private P staging

  const int tid  = threadIdx.x;
  const int wave = tid >> 5;
  const int lane = tid & 31;
  const int ln   = lane & 15;
  const int grp  = lane >> 4;
  const int h = blockIdx.y, b = blockIdx.z;
  const int q0 = blockIdx.x * AQB;
  const int qrow0 = q0 + wave * 16;

  // zero_factor = clip(softplus(factor), 1e-5, 0.1)
  float zf = logf(1.0f + __expf(factor[0]));
  zf = fminf(fmaxf(zf, 1e-5f), 0.1f);
  const float scale = 0.125f;                // hd^-0.5

  // stage Q block once: 128x64 = 2048 float4, 8 per thread (async path)
#pragma unroll
  for (int i = 0; i < 8; ++i) {
    int flat = i * 256 + tid;
    int r = flat >> 4, c4 = flat & 15;
    stage_b128(&Q[((size_t)(b * SEQ + q0 + r)) * DIM + h * HD + c4 * 4], &Qs[r][c4 * 4]);
  }

  float mrow[8], lrow[8];
#pragma unroll
  for (int r = 0; r < 8; ++r) { mrow[r] = -1e30f; lrow[r] = 0.0f; }
  v8f o[4];
#pragma unroll
  for (int ni = 0; ni < 4; ++ni) o[ni] = (v8f){};

  const size_t qk_base = ((size_t)(b * HEADS + h)) * SEQ * SEQ;

  for (int j = 0; j < SEQ / AKB; ++j) {
    __syncthreads();
    // stage K (transposed, through VGPRs) and V (straight copy -> async)
#pragma unroll
    for (int i = 0; i < 2; ++i) {
      int flat = i * 256 + tid;
      int n = flat >> 4, c4 = flat & 15;
      size_t g = ((size_t)(b * SEQ + j * AKB + n)) * DIM + h * HD + c4 * 4;
      float4 kv = *(const float4*)&K[g];
      Ks[c4 * 4 + 0][n] = kv.x; Ks[c4 * 4 + 1][n] = kv.y;
      Ks[c4 * 4 + 2][n] = kv.z; Ks[c4 * 4 + 3][n] = kv.w;
      stage_b128(&V[g], &Vs[n][c4 * 4]);
    }
    stage_wait();
    __syncthreads();

    // ---- S = Q K^T (wave-uniform skip keeps EXEC full for WMMA) ----
    v8f s0 = {}, s1 = {};
    if (j * AKB <= qrow0 + 15) {
#pragma unroll
      for (int kk = 0; kk < HD; kk += 4) {
        v2f a, b0, b1;
        a.x  = Qs[wave * 16 + ln][kk + 2 * grp];
        a.y  = Qs[wave * 16 + ln][kk + 2 * grp + 1];
        b0.x = Ks[kk + 2 * grp][ln];       b0.y = Ks[kk + 2 * grp + 1][ln];
        b1.x = Ks[kk + 2 * grp][16 + ln];  b1.y = Ks[kk + 2 * grp + 1][16 + ln];
        s0 = __builtin_amdgcn_wmma_f32_16x16x4_f32(false, a, false, b0, (short)0, s0, false, false);
        s1 = __builtin_amdgcn_wmma_f32_16x16x4_f32(false, a, false, b1, (short)0, s1, false, false);
      }
    }

    // ---- mask (multiplicative; masked logits are 0.0 and remain in the
    //      softmax), zero-token scaling, logit writeout ----
    float tok0 = Ks[0][ln], tok1 = Ks[0][16 + ln];
    float zm0 = (tok0 == 0.0f) ? zf : 1.0f;
    float zm1 = (tok1 == 0.0f) ? zf : 1.0f;
    float p0[8], p1[8];
#pragma unroll
    for (int r = 0; r < 8; ++r) {
      int row = qrow0 + grp * 8 + r;
      int c0 = j * AKB + ln, c1 = c0 + 16;
      float v0 = (c0 <= row) ? s0[r] * scale * zm0 : 0.0f;
      float v1 = (c1 <= row) ? s1[r] * scale * zm1 : 0.0f;
      p0[r] = v0; p1[r] = v1;
      size_t rb = qk_base + (size_t)row * SEQ;
      qk_out[rb + c0] = v0;
      qk_out[rb + c1] = v1;
    }

    // ---- wave-local online softmax ----
#pragma unroll
    for (int r = 0; r < 8; ++r) {
      float mv = fmaxf(p0[r], p1[r]);
#pragma unroll
      for (int off = 8; off >= 1; off >>= 1)
        mv = fmaxf(mv, __shfl_xor(mv, off, 32));
      float mi    = fmaxf(mrow[r], mv);
      float alpha = __expf(mrow[r] - mi);
      mrow[r] = mi;
      float e0 = __expf(p0[r] - mi);
      float e1 = __expf(p1[r] - mi);
      float ls = e0 + e1;
#pragma unroll
      for (int off = 8; off >= 1; off >>= 1)
        ls += __shfl_xor(ls, off, 32);
      lrow[r] = lrow[r] * alpha + ls;
      o[0][r] *= alpha; o[1][r] *= alpha; o[2][r] *= alpha; o[3][r] *= alpha;
      Ps[wave][grp * 8 + r][ln]      = e0;   // C layout -> LDS for A reload
      Ps[wave][grp * 8 + r][16 + ln] = e1;
    }

    // ---- O += P(16x32) * V(32x64): A fragment reused across 4 N tiles ----
#pragma unroll
    for (int kk = 0; kk < AKB; kk += 4) {
      v2f a;
      a.x = Ps[wave][ln][kk + 2 * grp];
      a.y = Ps[wave][ln][kk + 2 * grp + 1];
#pragma unroll
      for (int ni = 0; ni < 4; ++ni) {
        v2f bv;
        bv.x = Vs[kk + 2 * grp][ni * 16 + ln];
        bv.y = Vs[kk + 2 * grp + 1][ni * 16 + ln];
        o[ni] = __builtin_amdgcn_wmma_f32_16x16x4_f32(false, a, false, bv, (short)0, o[ni], false, false);
      }
    }
  }

  // finalize: divide by softmax denom, write attn in (B,S,D) layout
#pragma unroll
  for (int r = 0; r < 8; ++r) {
    float inv = 1.0f / lrow[r];
    int row = qrow0 + grp * 8 + r;
#pragma unroll
    for (int ni = 0; ni < 4; ++ni)
      attn_out[((size_t)(b * SEQ + row)) * DIM + h * HD + ni * 16 + ln] = o[ni][r] * inv;
  }
}

// ================================ launch ===================================
extern "C" void kernel_launch(void* const* d_in, const int* in_sizes, int n_in,
                              void* d_out, int out_size, void* d_ws, size_t ws_size,
                              hipStream_t stream) {
  const float* x      = (const float*)d_in[0];
  const float* Wq     = (const float*)d_in[1];
  const float* bq     = (const float*)d_in[2];
  const float* Wk     = (const float*)d_in[3];
  const float* Wv     = (const float*)d_in[4];
  const float* bv     = (const float*)d_in[5];
  const float* Wo     = (const float*)d_in[6];
  const float* bo     = (const float*)d_in[7];
  const float* factor = (const float*)d_in[8];

  const size_t nBSD = (size_t)BATCH * SEQ * DIM;     // 4,194,304
  float* out = (float*)d_out;
  float* qk  = out + nBSD;                           // (B,H,S,S) fp32

  float* q    = (float*)d_ws;
  float* k    = q + nBSD;
  float* v    = k + nBSD;
  float* attn = v + nBSD;

  const int M = BATCH * SEQ;                         // 4096
  dim3 gGemm(DIM / GN, M / GM);                      // (16, 32)

  gemm_xwt_kernel<<<gGemm, 256, 0, stream>>>(x, Wq, bq,      q, M, DIM, DIM);
  gemm_xwt_kernel<<<gGemm, 256, 0, stream>>>(x, Wk, nullptr, k, M, DIM, DIM);
  gemm_xwt_kernel<<<gGemm, 256, 0, stream>>>(x, Wv, bv,      v, M, DIM, DIM);

  const int pairs = BATCH * SEQ * HEADS * (HD / 2);  // 2,097,152
  rope_kernel<<<pairs / 256, 256, 0, stream>>>(q, pairs);
  rope_kernel<<<pairs / 256, 256, 0, stream>>>(k, pairs);

  attn_kernel<<<dim3(SEQ / AQB, HEADS, BATCH), 256, 0, stream>>>(q, k, v, factor, qk, attn);

  gemm_xwt_kernel<<<gGemm, 256, 0, stream>>>(attn, Wo, bo, out, M, DIM, DIM);
}